// VectorQuantizer_45904610460112
// MI455X (gfx1250) — compile-verified
//
#include <hip/hip_runtime.h>

typedef __attribute__((ext_vector_type(2))) float v2f;
typedef __attribute__((ext_vector_type(8))) float v8f;

#define VQ_B   32
#define VQ_D   256
#define VQ_HW  1024
#define VQ_N   (VQ_B * VQ_HW)        // 32768 rows
#define VQ_K   1024                  // codebook entries
#define ZQ_ELEMS (VQ_N * VQ_D)       // 8388608
#define LOSS_OFF ZQ_ELEMS
#define IDX_OFF  (ZQ_ELEMS + 1)

// ws layout (floats): [0..127] block partial sums, [128..1151] cnorm
#define WS_PARTIALS 0
#define WS_CNORM    128

// ---------------------------------------------------------------------------
// Kernel 0: per-code squared norms ||c_k||^2
// ---------------------------------------------------------------------------
__global__ __launch_bounds__(256) void vq_cnorm_kernel(
    const float* __restrict__ cb, float* __restrict__ cnorm) {
  int k = blockIdx.x * 256 + threadIdx.x;       // 4 blocks x 256 = 1024
  const float4* row = (const float4*)(cb + (size_t)k * VQ_D);
  float s = 0.f;
#pragma unroll 8
  for (int i = 0; i < VQ_D / 4; ++i) {
    float4 v = row[i];
    s += v.x * v.x + v.y * v.y + v.z * v.z + v.w * v.w;
  }
  cnorm[k] = s;
}

// ---------------------------------------------------------------------------
// Kernel 1: distances via V_WMMA_F32_16X16X4_F32 + fused argmin.
// Each wave: one 16-row strip of zf (A matrix held in 128 VGPRs).
// Each workgroup (8 waves): loops 64 codebook column tiles staged in LDS.
// 4 independent accumulators break the WMMA RAW chain (latency hiding).
// Score s = ||c||^2 - 2*z.c  (||z||^2 dropped: row-constant for argmin).
// ---------------------------------------------------------------------------
__global__ __launch_bounds__(256, 1) void vq_argmin_kernel(
    const float* __restrict__ z, const float* __restrict__ cb,
    const float* __restrict__ cnorm, float* __restrict__ idx_out) {
  __shared__ float lds[16 * 260];               // 16 cols x 256 dims, pad->260
  const int t    = threadIdx.x;
  const int wave = t >> 5;
  const int lane = t & 31;
  const int hi   = lane >> 4;                   // half-wave select
  const int lo   = lane & 15;

  const int strip = blockIdx.x * 8 + wave;      // 2048 strips of 16 rows
  const int base  = strip * 16;
  const int b     = base >> 10;                 // batch index (HW=1024)
  const int hw    = (base & 1023) + lo;         // row within batch, per lane

  // --- Preload A strip: zf[base+lo][*]. ISA A-layout for f32 16x4:
  // lanes 0-15: VGPR0=K(4s), VGPR1=K(4s+1); lanes 16-31: K(4s+2),K(4s+3).
  const float* zb = z + (size_t)b * VQ_D * VQ_HW + hw;
  v2f a[64];
#pragma unroll
  for (int s = 0; s < 64; ++s) {
    int kd = s * 4 + 2 * hi;
    a[s].x = zb[(size_t)kd * VQ_HW];
    a[s].y = zb[(size_t)(kd + 1) * VQ_HW];
  }

  float bestd[8];
  int   bestk[8];
#pragma unroll
  for (int i = 0; i < 8; ++i) { bestd[i] = 3.4e38f; bestk[i] = 0; }

  for (int ct = 0; ct < 64; ++ct) {
    __syncthreads();
    // Stage 16x256 codebook tile into LDS (contiguous 4 KB x 4 iters).
    const float4* src = (const float4*)(cb + (size_t)ct * 16 * VQ_D);
#pragma unroll
    for (int it = 0; it < 4; ++it) {
      int f   = it * 256 + t;                   // float4 index 0..1023
      float4 v = src[f];
      int col = f >> 6;
      int dim = (f & 63) * 4;
      *(float4*)&lds[col * 260 + dim] = v;
    }
    if (ct < 63)
      __builtin_prefetch(cb + (size_t)(ct + 1) * 16 * VQ_D, 0, 3);
    __syncthreads();

    // B-matrix 4x16 f32 layout mirrors A: lane supplies column (lo),
    // K pair selected by half-wave.  Padded stride 260 -> conflict-free.
    v8f acc[4];
#pragma unroll
    for (int j = 0; j < 4; ++j)
      acc[j] = (v8f){0.f, 0.f, 0.f, 0.f, 0.f, 0.f, 0.f, 0.f};

    const float* brow = &lds[lo * 260 + 2 * hi];
#pragma unroll
    for (int s = 0; s < 64; ++s) {
      v2f bv = *(const v2f*)(brow + s * 4);
      acc[s & 3] = __builtin_amdgcn_wmma_f32_16x16x4_f32(
          false, a[s], false, bv, (short)0, acc[s & 3], false, false);
    }

    // C layout: VGPR i -> row M = i + 8*hi, column N = lo.
    int   col = ct * 16 + lo;
    float cn  = cnorm[col];
#pragma unroll
    for (int i = 0; i < 8; ++i) {
      float dot = (acc[0][i] + acc[1][i]) + (acc[2][i] + acc[3][i]);
      float d = cn - 2.0f * dot;
      bool better = (d < bestd[i]) || (d == bestd[i] && col < bestk[i]);
      if (better) { bestd[i] = d; bestk[i] = col; }
    }
  }

  // Butterfly (val,idx) min across the 16 lanes of each half-wave.
#pragma unroll
  for (int i = 0; i < 8; ++i) {
#pragma unroll
    for (int off = 1; off < 16; off <<= 1) {
      float od = __shfl_xor(bestd[i], off, 32);
      int   ok = __shfl_xor(bestk[i], off, 32);
      bool better = (od < bestd[i]) || (od == bestd[i] && ok < bestk[i]);
      if (better) { bestd[i] = od; bestk[i] = ok; }
    }
  }
  if (lo == 0) {
#pragma unroll
    for (int i = 0; i < 8; ++i) {
      int row = base + hi * 8 + i;
      idx_out[row] = (float)bestk[i];           // exact for idx < 2^24
    }
  }
}

// ---------------------------------------------------------------------------
// Kernel 2: gather z_q = codebook[idx] into NCHW + per-block SSE partials.
// Coalesced over hw (contiguous per c-slice); codebook rows hit L2.
// ---------------------------------------------------------------------------
__global__ __launch_bounds__(256) void vq_gather_kernel(
    const float* __restrict__ z, const float* __restrict__ cb,
    const float* __restrict__ idxf, float* __restrict__ out,
    float* __restrict__ partials) {
  __shared__ float red[256];
  const int t  = threadIdx.x;
  const int n  = blockIdx.x * 256 + t;          // 128 blocks cover N=32768
  const int b  = n >> 10;
  const int hw = n & 1023;
  const int idx = (int)idxf[n];

  const float* crow = cb + (size_t)idx * VQ_D;
  const float* zp   = z   + (size_t)b * VQ_D * VQ_HW + hw;
  float*       op   = out + (size_t)b * VQ_D * VQ_HW + hw;

  float sum = 0.f;
#pragma unroll 4
  for (int c = 0; c < VQ_D; ++c) {
    float v  = crow[c];
    float zv = zp[(size_t)c * VQ_HW];
    op[(size_t)c * VQ_HW] = v;                  // z_q_st == z_q numerically
    float d = v - zv;
    sum += d * d;
  }

  red[t] = sum;
  __syncthreads();
  for (int sdown = 128; sdown > 0; sdown >>= 1) {
    if (t < sdown) red[t] += red[t + sdown];
    __syncthreads();
  }
  if (t == 0) partials[blockIdx.x] = red[0];
}

// ---------------------------------------------------------------------------
// Kernel 3: vq_loss = (1 + 0.25) * SSE / numel   (deterministic final sum)
// ---------------------------------------------------------------------------
__global__ void vq_loss_kernel(const float* __restrict__ partials,
                               float* __restrict__ loss_out) {
  if (threadIdx.x == 0 && blockIdx.x == 0) {
    float s = 0.f;
    for (int i = 0; i < 128; ++i) s += partials[i];
    loss_out[0] = 1.25f * s / (float)ZQ_ELEMS;
  }
}

// ---------------------------------------------------------------------------
extern "C" void kernel_launch(void* const* d_in, const int* in_sizes, int n_in,
                              void* d_out, int out_size, void* d_ws,
                              size_t ws_size, hipStream_t stream) {
  const float* z  = (const float*)d_in[0];   // [32,256,32,32]
  const float* cb = (const float*)d_in[1];   // [1024,256]
  float* out      = (float*)d_out;           // [zq | loss | indices]
  float* wsf      = (float*)d_ws;
  float* partials = wsf + WS_PARTIALS;
  float* cnorm    = wsf + WS_CNORM;
  float* idxf     = out + IDX_OFF;

  vq_cnorm_kernel <<<4,   256, 0, stream>>>(cb, cnorm);
  vq_argmin_kernel<<<256, 256, 0, stream>>>(z, cb, cnorm, idxf);
  vq_gather_kernel<<<128, 256, 0, stream>>>(z, cb, idxf, out, partials);
  vq_loss_kernel  <<<1,    32, 0, stream>>>(partials, out + LOSS_OFF);
}